// DynamicConvBlock_84018150244709
// MI455X (gfx1250) — compile-verified
//
#include <hip/hip_runtime.h>
#include <math.h>

// ---------------------------------------------------------------------------
// DynamicConvBlock for MI455X (gfx1250):
//   stage 1: 1x1 conv (GEMM 384x384 @ 384x65536) via V_WMMA_F32_16X16X4_F32
//            wave tile 64x64 (16 accumulators), block tile 128x256
//            + BN fold + residual  -> x_pw (workspace)
//   stage 2: 13x13 depthwise conv + BN + 1x1-depthwise branch + exact GELU
// ---------------------------------------------------------------------------

typedef __attribute__((ext_vector_type(2))) float fv2;
typedef __attribute__((ext_vector_type(8))) float fv8;

#define C_CH 384
#define HW   4096      // 64*64
#define EPS  1e-5f

__device__ __forceinline__ fv8 wmma4(fv2 a, fv2 b, fv8 c) {
  // D = A(16x4 f32) * B(4x16 f32) + C(16x16 f32)
  return __builtin_amdgcn_wmma_f32_16x16x4_f32(
      /*neg_a=*/false, a, /*neg_b=*/false, b,
      /*c_mod=*/(short)0, c, /*reuse_a=*/false, /*reuse_b=*/false);
}

// ---------------------------------------------------------------------------
// Prep: fold BN scale into pw_w, transpose into WMMA-A layout.
// Wt2[(k>>1)*768 + m*2 + (k&1)] = pw_w[m][k] * scale[m]
// so a lane's A operand {W[m,k],W[m,k+1]} is one contiguous 8B read,
// coalesced across lanes in m.
// ---------------------------------------------------------------------------
__global__ void prep_kernel(const float* __restrict__ pw_w,
                            const float* __restrict__ pw_g,
                            const float* __restrict__ pw_b,
                            const float* __restrict__ pw_m,
                            const float* __restrict__ pw_v,
                            float* __restrict__ Wt2,
                            float* __restrict__ shift) {
  int idx = blockIdx.x * 256 + threadIdx.x;
  if (idx >= C_CH * C_CH) return;
  int o = idx / C_CH, c = idx % C_CH;
  float sc = pw_g[o] * rsqrtf(pw_v[o] + EPS);
  Wt2[(size_t)(c >> 1) * (2 * C_CH) + o * 2 + (c & 1)] = pw_w[idx] * sc;
  if (c == 0) shift[o] = pw_b[o] - pw_m[o] * sc;
}

// ---------------------------------------------------------------------------
// Stage 1: x_pw[b,m,n] = sum_k W[m,k]*x[b,k,n]*scale[m] + shift[m] + x[b,m,n]
// Block = 256 thr (8 wave32s) arranged 2(M) x 4(N); wave tile 64x64.
// Per k-step: 4 b64 A-loads + 8 b32 B-loads feed 16 v_wmma_f32_16x16x4_f32.
// ---------------------------------------------------------------------------
__global__ __launch_bounds__(256) void pw_gemm_kernel(
    const float* __restrict__ x, const float* __restrict__ Wt2,
    const float* __restrict__ shift, float* __restrict__ xpw) {
  const int lane = threadIdx.x & 31;
  const int wave = threadIdx.x >> 5;
  const int lh   = lane & 15;
  const int khi  = (lane >> 4) << 1;  // 0 (lanes 0-15) or 2 (lanes 16-31)
  const int mrow = (lane >> 4) << 3;  // C/D rows: +0 or +8
  const int mb   = blockIdx.y * 128 + (wave & 1) * 64;
  const int nb   = blockIdx.x * 256 + (wave >> 1) * 64;
  const int b    = blockIdx.z;
  const float* xb = x + (size_t)b * C_CH * HW;

  fv8 acc[4][4];
#pragma unroll
  for (int mt = 0; mt < 4; ++mt)
#pragma unroll
    for (int nt = 0; nt < 4; ++nt) acc[mt][nt] = fv8{};

  // Per-lane operand column indices.
  int moff[4], nidx[4];
#pragma unroll
  for (int t = 0; t < 4; ++t) {
    moff[t] = 2 * (mb + t * 16 + lh);  // float offset into a Wt2 pair-row
    nidx[t] = nb + t * 16 + lh;
  }

  for (int k = 0; k < C_CH; k += 4) {
    const int kk = k + khi;                                   // even
    const float* wp  = Wt2 + (size_t)(kk >> 1) * (2 * C_CH);  // k-pair row
    const float* xk0 = xb + (size_t)kk * HW;
    const float* xk1 = xk0 + HW;

    fv2 a[4];
#pragma unroll
    for (int mt = 0; mt < 4; ++mt) a[mt] = *(const fv2*)(wp + moff[mt]);

    fv2 bb[4];
#pragma unroll
    for (int nt = 0; nt < 4; ++nt) {
      bb[nt].x = xk0[nidx[nt]];
      bb[nt].y = xk1[nidx[nt]];
    }

#pragma unroll
    for (int mt = 0; mt < 4; ++mt)
#pragma unroll
      for (int nt = 0; nt < 4; ++nt)
        acc[mt][nt] = wmma4(a[mt], bb[nt], acc[mt][nt]);
  }

  // Epilogue: C/D layout — VGPR r, lanes 0-15 -> M=r, lanes 16-31 -> M=r+8.
#pragma unroll
  for (int mt = 0; mt < 4; ++mt) {
#pragma unroll
    for (int r = 0; r < 8; ++r) {
      const int m = mb + mt * 16 + mrow + r;
      const float sh = shift[m];
      const float* xrow = xb + (size_t)m * HW;
      float* orow = xpw + ((size_t)b * C_CH + m) * HW;
#pragma unroll
      for (int nt = 0; nt < 4; ++nt) {
        const int n = nidx[nt];
        orow[n] = acc[mt][nt][r] + sh + xrow[n];
      }
    }
  }
}

// ---------------------------------------------------------------------------
// Stage 2: depthwise 13x13 + BN + per-channel branch + exact GELU.
// ---------------------------------------------------------------------------
#define TILE 32
#define HALO 6
#define ITD  44   // 32 + 2*6
#define ISTR 48   // padded LDS row stride (16B-aligned float4 rows)

__device__ __forceinline__ float gelu_exact(float v) {
  return 0.5f * v * (1.0f + erff(v * 0.70710678118654752440f));
}

__global__ __launch_bounds__(256) void dwconv_kernel(
    const float* __restrict__ xpw,
    const float* __restrict__ dwk_w,
    const float* __restrict__ dwk_g, const float* __restrict__ dwk_b,
    const float* __restrict__ dwk_m, const float* __restrict__ dwk_v,
    const float* __restrict__ dw1_w,
    const float* __restrict__ dw1_g, const float* __restrict__ dw1_b,
    const float* __restrict__ dw1_m, const float* __restrict__ dw1_v,
    float* __restrict__ out) {
  __shared__ float s_in[ITD * ISTR];   // 44*48*4 = 8448 B
  __shared__ float s_w[176];
  const int c  = blockIdx.y;
  const int b  = blockIdx.z;
  const int h0 = (blockIdx.x >> 1) * TILE;
  const int w0 = (blockIdx.x & 1) * TILE;
  const int tid = threadIdx.x;
  const float* src = xpw + ((size_t)b * C_CH + c) * HW;

  for (int i = tid; i < 169; i += 256) s_w[i] = dwk_w[c * 169 + i];
  for (int i = tid; i < ITD * ITD; i += 256) {
    int r = i / ITD, q = i - r * ITD;
    int h = h0 - HALO + r, w = w0 - HALO + q;
    float v = 0.0f;
    if ((unsigned)h < 64u && (unsigned)w < 64u) v = src[h * 64 + w];
    s_in[r * ISTR + q] = v;
  }
  __syncthreads();

  const float sck = dwk_g[c] * rsqrtf(dwk_v[c] + EPS);
  const float shk = dwk_b[c] - dwk_m[c] * sck;
  const float sc1 = dw1_g[c] * rsqrtf(dw1_v[c] + EPS);
  const float s1  = dw1_w[c] * sc1;
  const float sh1 = dw1_b[c] - dw1_m[c] * sc1;

  const int th = tid >> 3;         // 0..31  (output row)
  const int tw = (tid & 7) << 2;   // 0,4,...,28 (4 outputs along W)

  float a0 = 0.f, a1 = 0.f, a2 = 0.f, a3 = 0.f;
  for (int i = 0; i < 13; ++i) {
    const float* row = &s_in[(th + i) * ISTR + tw];   // 16B aligned
    float4 p0 = *(const float4*)(row + 0);
    float4 p1 = *(const float4*)(row + 4);
    float4 p2 = *(const float4*)(row + 8);
    float4 p3 = *(const float4*)(row + 12);
    float v[16] = { p0.x, p0.y, p0.z, p0.w,
                    p1.x, p1.y, p1.z, p1.w,
                    p2.x, p2.y, p2.z, p2.w,
                    p3.x, p3.y, p3.z, p3.w };
#pragma unroll
    for (int j = 0; j < 13; ++j) {
      const float wv = s_w[i * 13 + j];
      a0 = fmaf(wv, v[j],     a0);
      a1 = fmaf(wv, v[j + 1], a1);
      a2 = fmaf(wv, v[j + 2], a2);
      a3 = fmaf(wv, v[j + 3], a3);
    }
  }

  const float* ctr = &s_in[(th + HALO) * ISTR + (tw + HALO)];
  float4 o4;
  o4.x = gelu_exact(fmaf(a0, sck, shk) + fmaf(ctr[0], s1, sh1));
  o4.y = gelu_exact(fmaf(a1, sck, shk) + fmaf(ctr[1], s1, sh1));
  o4.z = gelu_exact(fmaf(a2, sck, shk) + fmaf(ctr[2], s1, sh1));
  o4.w = gelu_exact(fmaf(a3, sck, shk) + fmaf(ctr[3], s1, sh1));

  float* dst = out + ((size_t)b * C_CH + c) * HW + (h0 + th) * 64 + (w0 + tw);
  *(float4*)dst = o4;
}

// ---------------------------------------------------------------------------
extern "C" void kernel_launch(void* const* d_in, const int* in_sizes, int n_in,
                              void* d_out, int out_size, void* d_ws, size_t ws_size,
                              hipStream_t stream) {
  const float* x     = (const float*)d_in[0];
  const float* pw_w  = (const float*)d_in[1];
  const float* pw_g  = (const float*)d_in[2];
  const float* pw_b  = (const float*)d_in[3];
  const float* pw_m  = (const float*)d_in[4];
  const float* pw_v  = (const float*)d_in[5];
  const float* dwk_w = (const float*)d_in[6];
  const float* dwk_g = (const float*)d_in[7];
  const float* dwk_b = (const float*)d_in[8];
  const float* dwk_m = (const float*)d_in[9];
  const float* dwk_v = (const float*)d_in[10];
  const float* dw1_w = (const float*)d_in[11];
  const float* dw1_g = (const float*)d_in[12];
  const float* dw1_b = (const float*)d_in[13];
  const float* dw1_m = (const float*)d_in[14];
  const float* dw1_v = (const float*)d_in[15];
  float* out = (float*)d_out;

  // Workspace layout (floats): Wt2[147456] | shift[384 (+pad to 512)] | x_pw[25165824]
  float* Wt2   = (float*)d_ws;
  float* shift = Wt2 + C_CH * C_CH;
  float* xpw   = shift + 512;

  prep_kernel<<<dim3((C_CH * C_CH + 255) / 256), dim3(256), 0, stream>>>(
      pw_w, pw_g, pw_b, pw_m, pw_v, Wt2, shift);

  // Block tile 128(M) x 256(N): grid = (4096/256, 384/128, B) = (16, 3, 16)
  pw_gemm_kernel<<<dim3(HW / 256, C_CH / 128, 16), dim3(256), 0, stream>>>(
      x, Wt2, shift, xpw);

  dwconv_kernel<<<dim3(4, C_CH, 16), dim3(256), 0, stream>>>(
      xpw, dwk_w, dwk_g, dwk_b, dwk_m, dwk_v,
      dw1_w, dw1_g, dw1_b, dw1_m, dw1_v, out);
}